// GhostTraceGNN_38345468019206
// MI455X (gfx1250) — compile-verified
//
#include <hip/hip_runtime.h>
#include <hip/hip_bf16.h>

// ---------------------------------------------------------------------------
// GATv2 (3 layers) + graph pooling for MI455X (gfx1250, wave32).
// GEMMs use V_WMMA_F32_16X16X4_F32 with A-fragment reuse across CT col-tiles,
// batched B-fragment loads (clause + staggered s_wait_loadcnt), and a packed
// W layout so each B fragment is a single b64 load on a constant-stride ptr.
// Edge phase: L2-resident gathers + float atomics (segment softmax/aggregate).
// ---------------------------------------------------------------------------

typedef __attribute__((ext_vector_type(2))) float v2f;
typedef __attribute__((ext_vector_type(8))) float v8f;

#define NEG_SLOPE 0.2f
#define NEG_INF   (-3.0e38f)

// ---------------------------------------------------------------------------
// Pack W[K,M] row-major into Wpack[K/2][M] of float2:
//   Wpack[k2*M + c] = { W[(2*k2)*M + c], W[(2*k2+1)*M + c] }
// so a WMMA B fragment (two consecutive K rows, one column) is one b64 load.
// ---------------------------------------------------------------------------
__global__ void ghost_pack_w(const float* __restrict__ W,
                             float2* __restrict__ Wpack,
                             int K, int M) {
    int idx = blockIdx.x * blockDim.x + threadIdx.x;
    int total = (K / 2) * M;
    if (idx >= total) return;
    int k2 = idx / M;
    int c  = idx - k2 * M;
    float2 p;
    p.x = W[(size_t)(2 * k2) * M + c];
    p.y = W[(size_t)(2 * k2 + 1) * M + c];
    Wpack[idx] = p;
}

// ---------------------------------------------------------------------------
// Y[N,M] = X[N,K] @ W[K,M] + bias[M]   (W pre-packed as above)
// One wave -> one 16-row x (CT*16)-col stripe of Y: one A fragment feeds CT
// WMMAs per k-step. All CT B loads issue before the CT WMMAs so the compiler
// can clause them and stagger the loadcnt waits.
// Requires N%16==0, M%(16*CT)==0, K%4==0 (true: N=50000, M in {128,32}).
// Fragment layouts per CDNA5 ISA 7.12.2:
//   A 16x4 f32 : lanes 0-15 row M=l hold K={k,k+1}; lanes 16-31 hold K={k+2,k+3}
//   B 4x16 f32 : lanes 0-15 col N=l hold K rows {k,k+1}; lanes 16-31 {k+2,k+3}
//   C/D 16x16  : VGPR i, lanes 0-15 -> row r0+i, lanes 16-31 -> row r0+8+i
// ---------------------------------------------------------------------------
template <int CT>
__global__ void ghost_wmma_gemm_bias(const float* __restrict__ X,
                                     const float2* __restrict__ Wpack,
                                     const float* __restrict__ bias,
                                     float* __restrict__ Y,
                                     int Nrow, int K, int M) {
    const int lane = threadIdx.x & 31;
    const int wave = threadIdx.x >> 5;
    const int colGroups  = M / (16 * CT);
    const int totalTiles = (Nrow >> 4) * colGroups;
    const int tile = blockIdx.x * 4 + wave;
    if (tile >= totalTiles) return;          // wave-uniform: EXEC stays all-1s

    const int rt = tile / colGroups;
    const int cg = tile - rt * colGroups;
    const int r0 = rt << 4;
    const int c0 = cg * (16 * CT);
    const int half = lane >> 4;              // 0: K pair {k,k+1}, 1: {k+2,k+3}
    const int l    = lane & 15;
    const int ar   = r0 + l;                 // A row this lane carries

    // Constant-stride pointers: no per-iteration 64-bit multiplies.
    const v2f*    xp = (const v2f*)(X + (size_t)ar * K + 2 * half);
    const float2* wp = Wpack + (size_t)half * M + (c0 + l);
    const int kSteps = K >> 2;

    v8f acc[CT] = {};
    for (int ks = 0; ks < kSteps; ++ks) {
        v2f a = xp[0];
        v2f b[CT];
#pragma unroll
        for (int t = 0; t < CT; ++t) {       // all B loads first -> one clause
            float2 bb = wp[t * 16];
            b[t].x = bb.x;
            b[t].y = bb.y;
        }
#pragma unroll
        for (int t = 0; t < CT; ++t) {       // then the WMMAs, staggered waits
            acc[t] = __builtin_amdgcn_wmma_f32_16x16x4_f32(
                /*neg_a=*/false, a, /*neg_b=*/false, b[t],
                /*c_mod=*/(short)0, acc[t], /*reuse_a=*/false, /*reuse_b=*/false);
        }
        xp += 2;                             // +4 floats of K
        wp += 2 * M;                         // +2 packed K-rows
    }

#pragma unroll
    for (int t = 0; t < CT; ++t) {
        const int col = c0 + t * 16 + l;
        const float bv = bias[col];
#pragma unroll
        for (int i = 0; i < 8; ++i) {
            const int row = r0 + half * 8 + i;
            Y[(size_t)row * M + col] = acc[t][i] + bv;
        }
    }
}

// ---------------------------------------------------------------------------
// Per-layer init: smax=-inf, denom=0 over [N*H]; out=0 over [N*H*C]
// ---------------------------------------------------------------------------
__global__ void ghost_layer_init(float* __restrict__ smax,
                                 float* __restrict__ denom,
                                 float* __restrict__ out,
                                 size_t nH, size_t nHC) {
    size_t i = (size_t)blockIdx.x * blockDim.x + threadIdx.x;
    if (i < nH) { smax[i] = NEG_INF; denom[i] = 0.0f; }
    if (i < nHC) out[i] = 0.0f;
}

__device__ __forceinline__ void ghost_edge_endpoints(const int* __restrict__ ei,
                                                     long long e, int E,
                                                     int& s, int& d) {
    if (e < E) { s = ei[e]; d = ei[(size_t)E + e]; }
    else       { s = d = (int)(e - E); }     // appended self-loops
}

// ---------------------------------------------------------------------------
// score[e,h] = sum_c leakyrelu(xl[src][h,c] + xr[dst][h,c]) * att[h,c]
// and smax[dst,h] = max(...) via float atomic max.
// ---------------------------------------------------------------------------
__global__ void ghost_edge_score(const int* __restrict__ ei,
                                 const float* __restrict__ xl,
                                 const float* __restrict__ xr,
                                 const float* __restrict__ att,
                                 float* __restrict__ score,
                                 float* __restrict__ smax,
                                 int E, int Nn, int H, int C) {
    size_t idx = (size_t)blockIdx.x * blockDim.x + threadIdx.x;
    size_t total = (size_t)(E + Nn) * H;
    if (idx >= total) return;
    long long e = (long long)(idx / H);
    int h = (int)(idx % H);
    int s, d;
    ghost_edge_endpoints(ei, e, E, s, d);

    const float4* pl = (const float4*)(xl + ((size_t)s * H + h) * C);
    const float4* pr = (const float4*)(xr + ((size_t)d * H + h) * C);
    const float4* pa = (const float4*)(att + (size_t)h * C);
    float acc = 0.0f;
    for (int c4 = 0; c4 < (C >> 2); ++c4) {
        float4 vl = pl[c4], vr = pr[c4], va = pa[c4];
        float vx = vl.x + vr.x; vx = vx > 0.0f ? vx : NEG_SLOPE * vx;
        float vy = vl.y + vr.y; vy = vy > 0.0f ? vy : NEG_SLOPE * vy;
        float vz = vl.z + vr.z; vz = vz > 0.0f ? vz : NEG_SLOPE * vz;
        float vw = vl.w + vr.w; vw = vw > 0.0f ? vw : NEG_SLOPE * vw;
        acc += vx * va.x + vy * va.y + vz * va.z + vw * va.w;
    }
    score[idx] = acc;
    atomicMax(&smax[(size_t)d * H + h], acc);
}

// ---------------------------------------------------------------------------
// score[e,h] <- exp(score - smax[dst]);  denom[dst,h] += exp(...)
// ---------------------------------------------------------------------------
__global__ void ghost_edge_exp(const int* __restrict__ ei,
                               float* __restrict__ score,
                               const float* __restrict__ smax,
                               float* __restrict__ denom,
                               int E, int Nn, int H) {
    size_t idx = (size_t)blockIdx.x * blockDim.x + threadIdx.x;
    size_t total = (size_t)(E + Nn) * H;
    if (idx >= total) return;
    long long e = (long long)(idx / H);
    int h = (int)(idx % H);
    int s, d;
    ghost_edge_endpoints(ei, e, E, s, d);
    float ex = __expf(score[idx] - smax[(size_t)d * H + h]);
    score[idx] = ex;
    atomicAdd(&denom[(size_t)d * H + h], ex);
}

// ---------------------------------------------------------------------------
// score[e,h] <- ex / (denom[dst,h] + 1e-16)   (one divide per edge-head)
// ---------------------------------------------------------------------------
__global__ void ghost_edge_alpha(const int* __restrict__ ei,
                                 float* __restrict__ score,
                                 const float* __restrict__ denom,
                                 int E, int Nn, int H) {
    size_t idx = (size_t)blockIdx.x * blockDim.x + threadIdx.x;
    size_t total = (size_t)(E + Nn) * H;
    if (idx >= total) return;
    long long e = (long long)(idx / H);
    int h = (int)(idx % H);
    int s, d;
    ghost_edge_endpoints(ei, e, E, s, d);
    score[idx] = score[idx] / (denom[(size_t)d * H + h] + 1e-16f);
}

// ---------------------------------------------------------------------------
// out[dst, f..f+3] += alpha[e,h] * xl[src, f..f+3]   (float4 gather + 4 atomics)
// ---------------------------------------------------------------------------
__global__ void ghost_edge_aggregate(const int* __restrict__ ei,
                                     const float* __restrict__ alpha,
                                     const float* __restrict__ xl,
                                     float* __restrict__ out,
                                     int E, int Nn, int H, int C) {
    size_t idx = (size_t)blockIdx.x * blockDim.x + threadIdx.x;
    const int HC = H * C;
    const int Q  = HC >> 2;                  // float4 chunks per edge
    size_t total = (size_t)(E + Nn) * Q;
    if (idx >= total) return;
    long long e = (long long)(idx / Q);
    int q = (int)(idx % Q);
    int f = q << 2;
    int h = f / C;
    int s, d;
    ghost_edge_endpoints(ei, e, E, s, d);
    float a = alpha[(size_t)e * H + h];
    float4 v = *(const float4*)(xl + (size_t)s * HC + f);
    float* po = out + (size_t)d * HC + f;
    atomicAdd(po + 0, a * v.x);
    atomicAdd(po + 1, a * v.y);
    atomicAdd(po + 2, a * v.z);
    atomicAdd(po + 3, a * v.w);
}

// ---------------------------------------------------------------------------
// y = elu(out + bias)   (concat layers: HC = H*C; layer 3: H=1 so mean==out)
// ---------------------------------------------------------------------------
__global__ void ghost_finalize(const float* __restrict__ out,
                               const float* __restrict__ bias,
                               float* __restrict__ y,
                               size_t total, int M) {
    size_t i = (size_t)blockIdx.x * blockDim.x + threadIdx.x;
    if (i >= total) return;
    float v = out[i] + bias[i % M];
    y[i] = v > 0.0f ? v : (__expf(v) - 1.0f);
}

// ---------------------------------------------------------------------------
// Pooling: mean + max over batch segments (G=16, 32 feats)
// ---------------------------------------------------------------------------
__global__ void ghost_pool_init(float* __restrict__ psum,
                                float* __restrict__ pcnt,
                                float* __restrict__ pmax, int G, int C) {
    int i = blockIdx.x * blockDim.x + threadIdx.x;
    if (i < G * C) { psum[i] = 0.0f; pmax[i] = NEG_INF; }
    if (i < G) pcnt[i] = 0.0f;
}

__global__ void ghost_pool_acc(const float* __restrict__ hfeat,
                               const int* __restrict__ batch,
                               float* __restrict__ psum,
                               float* __restrict__ pcnt,
                               float* __restrict__ pmax,
                               int Nn, int C) {
    size_t idx = (size_t)blockIdx.x * blockDim.x + threadIdx.x;
    if (idx >= (size_t)Nn * C) return;
    int n = (int)(idx / C);
    int c = (int)(idx % C);
    int g = batch[n];
    float v = hfeat[idx];
    atomicAdd(&psum[g * C + c], v);
    atomicMax(&pmax[g * C + c], v);
    if (c == 0) atomicAdd(&pcnt[g], 1.0f);
}

__global__ void ghost_pool_final(const float* __restrict__ psum,
                                 const float* __restrict__ pcnt,
                                 const float* __restrict__ pmax,
                                 float* __restrict__ outp, int G, int C) {
    int i = blockIdx.x * blockDim.x + threadIdx.x;
    if (i >= G * 2 * C) return;
    int g = i / (2 * C);
    int c = i % (2 * C);
    outp[i] = (c < C) ? (psum[g * C + c] / pcnt[g])
                      : pmax[g * C + (c - C)];
}

// ---------------------------------------------------------------------------
// Host side
// ---------------------------------------------------------------------------
static inline int ghost_grid(size_t total, int block) {
    return (int)((total + block - 1) / block);
}

extern "C" void kernel_launch(void* const* d_in, const int* in_sizes, int n_in,
                              void* d_out, int out_size, void* d_ws, size_t ws_size,
                              hipStream_t stream) {
    const float* x     = (const float*)d_in[0];
    const int*   ei    = (const int*)d_in[1];    // edge_index [2,E]
    const int*   batch = (const int*)d_in[2];
    const float* Wl1 = (const float*)d_in[3];  const float* bl1 = (const float*)d_in[4];
    const float* Wr1 = (const float*)d_in[5];  const float* br1 = (const float*)d_in[6];
    const float* at1 = (const float*)d_in[7];  const float* b1  = (const float*)d_in[8];
    const float* Wl2 = (const float*)d_in[9];  const float* bl2 = (const float*)d_in[10];
    const float* Wr2 = (const float*)d_in[11]; const float* br2 = (const float*)d_in[12];
    const float* at2 = (const float*)d_in[13]; const float* b2  = (const float*)d_in[14];
    const float* Wl3 = (const float*)d_in[15]; const float* bl3 = (const float*)d_in[16];
    const float* Wr3 = (const float*)d_in[17]; const float* br3 = (const float*)d_in[18];
    const float* at3 = (const float*)d_in[19]; const float* b3  = (const float*)d_in[20];

    const int Nn = in_sizes[0] / 8;   // 50000
    const int Ee = in_sizes[1] / 2;   // 800000
    const int G  = 16;

    // Workspace carve-up (floats). All offsets multiples of 16 -> 16B aligned.
    float* ws = (float*)d_ws;
    size_t off = 0;
    float* bufXL  = ws + off; off += (size_t)Nn * 128;       // lin_l output
    float* bufXR  = ws + off; off += (size_t)Nn * 128;       // lin_r output
    float* bufAgg = ws + off; off += (size_t)Nn * 128;       // aggregated out
    float* bufH   = ws + off; off += (size_t)Nn * 128;       // layer activation
    float* score  = ws + off; off += (size_t)(Ee + Nn) * 2;  // score -> ex -> alpha
    float* smax   = ws + off; off += (size_t)Nn * 2;
    float* denom  = ws + off; off += (size_t)Nn * 2;
    float* wpackL = ws + off; off += (size_t)128 * 128;      // packed Wl (float2)
    float* wpackR = ws + off; off += (size_t)128 * 128;      // packed Wr (float2)
    float* psum   = ws + off; off += (size_t)G * 32;
    float* pcnt   = ws + off; off += (size_t)G;
    float* pmax   = ws + off; off += (size_t)G * 32;

    const int BLK = 256;

    auto run_layer = [&](const float* xin, int K,
                         const float* Wl, const float* bl,
                         const float* Wr, const float* br,
                         const float* att, const float* bias,
                         int H, int C) {
        const int M = H * C;
        const int packTotal = (K / 2) * M;
        ghost_pack_w<<<ghost_grid(packTotal, BLK), BLK, 0, stream>>>(Wl, (float2*)wpackL, K, M);
        ghost_pack_w<<<ghost_grid(packTotal, BLK), BLK, 0, stream>>>(Wr, (float2*)wpackR, K, M);

        if (M == 128) {                       // CT=4: wave covers 16x64
            const int totalTiles = (Nn / 16) * (M / 64);
            const int grid = (totalTiles + 3) / 4;
            ghost_wmma_gemm_bias<4><<<grid, 128, 0, stream>>>(xin, (const float2*)wpackL, bl, bufXL, Nn, K, M);
            ghost_wmma_gemm_bias<4><<<grid, 128, 0, stream>>>(xin, (const float2*)wpackR, br, bufXR, Nn, K, M);
        } else {                              // M=32, CT=2: wave covers 16x32
            const int totalTiles = (Nn / 16) * (M / 32);
            const int grid = (totalTiles + 3) / 4;
            ghost_wmma_gemm_bias<2><<<grid, 128, 0, stream>>>(xin, (const float2*)wpackL, bl, bufXL, Nn, K, M);
            ghost_wmma_gemm_bias<2><<<grid, 128, 0, stream>>>(xin, (const float2*)wpackR, br, bufXR, Nn, K, M);
        }

        const size_t nH  = (size_t)Nn * H;
        const size_t nHC = (size_t)Nn * M;
        ghost_layer_init<<<ghost_grid(nHC, BLK), BLK, 0, stream>>>(smax, denom, bufAgg, nH, nHC);

        const size_t eH = (size_t)(Ee + Nn) * H;
        const size_t eQ = (size_t)(Ee + Nn) * (M >> 2);
        ghost_edge_score<<<ghost_grid(eH, BLK), BLK, 0, stream>>>(
            ei, bufXL, bufXR, att, score, smax, Ee, Nn, H, C);
        ghost_edge_exp<<<ghost_grid(eH, BLK), BLK, 0, stream>>>(
            ei, score, smax, denom, Ee, Nn, H);
        ghost_edge_alpha<<<ghost_grid(eH, BLK), BLK, 0, stream>>>(
            ei, score, denom, Ee, Nn, H);
        ghost_edge_aggregate<<<ghost_grid(eQ, BLK), BLK, 0, stream>>>(
            ei, score, bufXL, bufAgg, Ee, Nn, H, C);
        ghost_finalize<<<ghost_grid(nHC, BLK), BLK, 0, stream>>>(
            bufAgg, bias, bufH, nHC, M);
    };

    run_layer(x,    8,   Wl1, bl1, Wr1, br1, at1, b1, 2, 64);   // -> bufH [N,128]
    run_layer(bufH, 128, Wl2, bl2, Wr2, br2, at2, b2, 2, 64);   // -> bufH [N,128]
    run_layer(bufH, 128, Wl3, bl3, Wr3, br3, at3, b3, 1, 32);   // -> bufH [N,32]

    ghost_pool_init<<<ghost_grid((size_t)G * 32, BLK), BLK, 0, stream>>>(psum, pcnt, pmax, G, 32);
    ghost_pool_acc<<<ghost_grid((size_t)Nn * 32, BLK), BLK, 0, stream>>>(
        bufH, batch, psum, pcnt, pmax, Nn, 32);
    ghost_pool_final<<<ghost_grid((size_t)G * 64, BLK), BLK, 0, stream>>>(
        psum, pcnt, pmax, (float*)d_out, G, 32);
}